// SpatialGroupEnhance_74560632258630
// MI455X (gfx1250) — compile-verified
//
#include <hip/hip_runtime.h>
#include <hip/hip_bf16.h>
#include <math.h>

typedef __attribute__((ext_vector_type(2))) float v2f;
typedef __attribute__((ext_vector_type(8))) float v8f;

#define B_  32
#define C_  256
#define HW_ 16384   // 128*128
#define NPIX (B_ * HW_)

// ---------------- K0: zero hist + sigmoid accumulator ----------------
__global__ void k_init(unsigned int* __restrict__ hist, float* __restrict__ sig) {
    hist[threadIdx.x] = 0u;
    if (threadIdx.x == 0) sig[0] = 0.0f;
}

// ---------------- K1: pooled[b,c] = spatial mean (one block per plane) ----------------
__global__ void k_pool(const float* __restrict__ x, float* __restrict__ pooled) {
    const int plane = blockIdx.x;                     // b*C + c
    const float4* p4 = (const float4*)(x + (size_t)plane * HW_);
    float s = 0.0f;
    #pragma unroll 4
    for (int i = threadIdx.x; i < HW_ / 4; i += 256) {
        float4 v = p4[i];
        s += (v.x + v.y) + (v.z + v.w);
    }
    __shared__ float red[256];
    red[threadIdx.x] = s;
    __syncthreads();
    for (int off = 128; off > 0; off >>= 1) {
        if (threadIdx.x < off) red[threadIdx.x] += red[threadIdx.x + off];
        __syncthreads();
    }
    if (threadIdx.x == 0) pooled[plane] = red[0] * (1.0f / (float)HW_);
}

// ---------------- K2: xn[b,p] via V_WMMA_F32_16X16X4_F32 ----------------
// Per wave: one 16-position spatial tile, K-loop over 256 channels in steps of 4.
// A (16x4 f32): pooled weights broadcast to all 16 rows.
//   lanes 0-15: a = {w[k0+0], w[k0+1]} ; lanes 16-31: a = {w[k0+2], w[k0+3]}
// B (4x16 f32): lanes 0-15: b = {x[k0+0,p], x[k0+1,p]} ; lanes 16-31: {x[k0+2,p], x[k0+3,p]}
// D: all rows identical; lane n (n<16), VGPR0 = dot(w, x[:,p0+n]).
__global__ void k_xn_wmma(const float* __restrict__ x,
                          const float* __restrict__ pooled,
                          float* __restrict__ xn) {
    __shared__ float wsh[C_];
    const int b = blockIdx.y;
    wsh[threadIdx.x] = pooled[b * C_ + threadIdx.x];
    __syncthreads();

    const int wave = threadIdx.x >> 5;
    const int lane = threadIdx.x & 31;
    const int half = lane >> 4;        // 0: K pair {0,1}, 1: K pair {2,3}
    const int n    = lane & 15;        // column within the 16-wide spatial tile
    const int p0   = (blockIdx.x * 8 + wave) * 16;

    const float* xb = x + (size_t)b * C_ * HW_ + (size_t)p0 + n;

    v8f acc = {};
    #pragma unroll 4
    for (int k0 = 0; k0 < C_; k0 += 4) {
        const int ka = k0 + half * 2;
        v2f a;
        a.x = wsh[ka];
        a.y = wsh[ka + 1];
        const float* px = xb + (size_t)ka * HW_;
        v2f bm;
        bm.x = px[0];
        bm.y = px[HW_];
        __builtin_prefetch(px + (size_t)4 * HW_, 0, 1);   // next K step -> global_prefetch_b8
        acc = __builtin_amdgcn_wmma_f32_16x16x4_f32(
                /*neg_a=*/false, a, /*neg_b=*/false, bm,
                /*c_mod=*/(short)0, acc, /*reuse_a=*/false, /*reuse_b=*/false);
    }
    if (half == 0)
        xn[(size_t)b * HW_ + p0 + n] = acc[0] * (1.0f / (float)C_);
}

// ---------------- K3: entro[p] = mean_b xn[b,p]; accumulate sigmoid sum ----------------
__global__ void k_entro_sig(const float* __restrict__ xn,
                            float* __restrict__ entro,
                            float* __restrict__ sig) {
    const int p = blockIdx.x * 256 + threadIdx.x;
    float e = 0.0f, s = 0.0f;
    #pragma unroll
    for (int b = 0; b < B_; ++b) {
        float v = xn[(size_t)b * HW_ + p];
        e += v;
        s += 1.0f / (1.0f + expf(-v));
    }
    entro[p] = e * (1.0f / (float)B_);
    __shared__ float red[256];
    red[threadIdx.x] = s;
    __syncthreads();
    for (int off = 128; off > 0; off >>= 1) {
        if (threadIdx.x < off) red[threadIdx.x] += red[threadIdx.x + off];
        __syncthreads();
    }
    if (threadIdx.x == 0) atomicAdd(sig, red[0]);
}

// ---------------- K4: min/max of entro (single block, deterministic) ----------------
__global__ void k_minmax(const float* __restrict__ entro, float* __restrict__ emm) {
    float mn = 3.402823466e38f, mx = -3.402823466e38f;
    for (int i = threadIdx.x; i < HW_; i += 256) {
        float v = entro[i];
        mn = fminf(mn, v);
        mx = fmaxf(mx, v);
    }
    __shared__ float rmn[256], rmx[256];
    rmn[threadIdx.x] = mn;
    rmx[threadIdx.x] = mx;
    __syncthreads();
    for (int off = 128; off > 0; off >>= 1) {
        if (threadIdx.x < off) {
            rmn[threadIdx.x] = fminf(rmn[threadIdx.x], rmn[threadIdx.x + off]);
            rmx[threadIdx.x] = fmaxf(rmx[threadIdx.x], rmx[threadIdx.x + off]);
        }
        __syncthreads();
    }
    if (threadIdx.x == 0) { emm[0] = rmn[0]; emm[1] = rmx[0]; }
}

// ---------------- K5: 256-bin histogram over normalized entro ----------------
// jnp.histogram(bins=256, range=(0,255)): edges i*255/256, left-inclusive,
// last bin right-inclusive  ==  floor(e * 256/255) clamped to [0,255].
__global__ void k_hist(const float* __restrict__ entro,
                       const float* __restrict__ emm,
                       unsigned int* __restrict__ hist) {
    __shared__ unsigned int h[256];
    h[threadIdx.x] = 0u;
    __syncthreads();
    const float mn = emm[0];
    const float sc = 255.0f / (emm[1] - mn);
    const int p = blockIdx.x * 256 + threadIdx.x;
    const float e = (entro[p] - mn) * sc;                 // in [0,255]
    int bin = (int)floorf(e * (256.0f / 255.0f));
    bin = bin < 0 ? 0 : (bin > 255 ? 255 : bin);
    atomicAdd(&h[bin], 1u);
    __syncthreads();
    atomicAdd(&hist[threadIdx.x], h[threadIdx.x]);
}

// ---------------- K6: entropy + final scalar ----------------
__global__ void k_final(const unsigned int* __restrict__ hist,
                        const float* __restrict__ sig,
                        float* __restrict__ out) {
    const int t = threadIdx.x;
    const float his = (float)hist[t] * (1.0f / (float)HW_);
    const float term = his * (-logf(his + 1e-8f));
    const float nz = hist[t] ? 1.0f : 0.0f;
    __shared__ float rt[256], rn[256];
    rt[t] = term;
    rn[t] = nz;
    __syncthreads();
    for (int off = 128; off > 0; off >>= 1) {
        if (t < off) { rt[t] += rt[t + off]; rn[t] += rn[t + off]; }
        __syncthreads();
    }
    if (t == 0) {
        const float entro_final = rt[0] / rn[0];
        const float s = sig[0] * (1.0f / (float)NPIX);
        out[0] = s + entro_final * 10.0f;
    }
}

extern "C" void kernel_launch(void* const* d_in, const int* in_sizes, int n_in,
                              void* d_out, int out_size, void* d_ws, size_t ws_size,
                              hipStream_t stream) {
    const float* x = (const float*)d_in[0];
    float* out = (float*)d_out;

    // workspace layout (floats)
    float* pooled = (float*)d_ws;                 // 8192
    float* xn     = pooled + B_ * C_;             // 524288
    float* entro  = xn + NPIX;                    // 16384
    float* sig    = entro + HW_;                  // 1
    float* emm    = sig + 1;                      // 2 (emin, emax)
    unsigned int* hist = (unsigned int*)(emm + 2);// 256

    k_init<<<1, 256, 0, stream>>>(hist, sig);
    k_pool<<<B_ * C_, 256, 0, stream>>>(x, pooled);
    {
        dim3 grid(HW_ / (16 * 8), B_);            // 128 x 32 blocks, 8 waves/block
        k_xn_wmma<<<grid, 256, 0, stream>>>(x, pooled, xn);
    }
    k_entro_sig<<<HW_ / 256, 256, 0, stream>>>(xn, entro, sig);
    k_minmax<<<1, 256, 0, stream>>>(entro, emm);
    k_hist<<<HW_ / 256, 256, 0, stream>>>(entro, emm, hist);
    k_final<<<1, 256, 0, stream>>>(hist, sig, out);
}